// ESE_86543591015146
// MI455X (gfx1250) — compile-verified
//
#include <hip/hip_runtime.h>

// Vector types for WMMA operands (fp32 path: 16x16x4, A/B = 2 VGPRs, C/D = 8 VGPRs)
typedef __attribute__((ext_vector_type(2))) float v2f;
typedef __attribute__((ext_vector_type(4))) float v4f;
typedef __attribute__((ext_vector_type(8))) float v8f;

#define B_DIM 32
#define C_DIM 768
#define HW 3136          // 56*56
#define HW4 784          // HW / 4

// ---------------------------------------------------------------------------
// Kernel 1: global average pool.  One 256-thread block per (b,c) row.
// x layout [B, C, H, W] row-major -> each (b,c) row is 3136 contiguous floats.
// ---------------------------------------------------------------------------
__global__ void ese_pool_kernel(const float* __restrict__ x,
                                float* __restrict__ xse) {
    const int row = blockIdx.x;                       // b*C + c
    const v4f* xr = (const v4f*)(x + (size_t)row * HW);

    float sum = 0.0f;
    for (int i = threadIdx.x; i < HW4; i += blockDim.x) {
        v4f v = xr[i];
        sum += (v.x + v.y) + (v.z + v.w);
    }

    __shared__ float red[256];
    red[threadIdx.x] = sum;
    __syncthreads();
    if (threadIdx.x < 128) red[threadIdx.x] += red[threadIdx.x + 128];
    __syncthreads();
    if (threadIdx.x < 64)  red[threadIdx.x] += red[threadIdx.x + 64];
    __syncthreads();
    if (threadIdx.x < 32) {
        float v = red[threadIdx.x] + red[threadIdx.x + 32];
        // wave32 shuffle reduction
        for (int off = 16; off > 0; off >>= 1)
            v += __shfl_down(v, off, 32);
        if (threadIdx.x == 0)
            xse[row] = v * (1.0f / (float)HW);
    }
}

// ---------------------------------------------------------------------------
// Kernel 2: dense (y = xse @ W^T + b) + hardsigmoid using V_WMMA_F32_16X16X4_F32.
// One wave (32 threads) per 16x16 output tile; EXEC is all-1s as WMMA requires.
//
// ISA VGPR layouts (CDNA5 7.12.2):
//   A 16x4 f32:  lane L: m = L%16; lanes 0-15 hold K={k0,k0+1}, lanes 16-31
//                hold K={k0+2,k0+3}  -> contiguous float2 of xse row.
//   B 4x16 f32:  lane L: n = L%16; same K split per half-wave.
//                B[k][n] = W[nTile*16+n][k]  -> contiguous float2 of W row.
//   C/D 16x16 f32: VGPR r: m = r + 8*(L>=16), n = L%16.
// ---------------------------------------------------------------------------
__global__ void ese_gemm_hsig_kernel(const float* __restrict__ xse,   // [32, 768]
                                     const float* __restrict__ Wm,    // [768, 768]
                                     const float* __restrict__ bias,  // [768]
                                     float* __restrict__ s) {         // [32, 768]
    const int lane  = threadIdx.x;        // 0..31
    const int ntile = blockIdx.x;         // 0..47
    const int mtile = blockIdx.y;         // 0..1
    const int half  = lane >> 4;          // 0 | 1
    const int id16  = lane & 15;

    const float* arow = xse + (size_t)(mtile * 16 + id16) * C_DIM;  // A row m
    const float* brow = Wm  + (size_t)(ntile * 16 + id16) * C_DIM;  // B col n

    v8f acc = {};
    #pragma unroll 4
    for (int k0 = 0; k0 < C_DIM; k0 += 4) {
        const int k = k0 + 2 * half;
        v2f a = *(const v2f*)(arow + k);
        v2f b = *(const v2f*)(brow + k);
        acc = __builtin_amdgcn_wmma_f32_16x16x4_f32(
            /*neg_a=*/false, a, /*neg_b=*/false, b,
            /*c_mod=*/(short)0, acc, /*reuse_a=*/false, /*reuse_b=*/false);
    }

    const int nn = ntile * 16 + id16;
    const float bb = bias[nn];
    #pragma unroll
    for (int r = 0; r < 8; ++r) {
        const int mm = mtile * 16 + r + 8 * half;
        const float y = acc[r] + bb;
        // hardsigmoid: clip(y/6 + 0.5, 0, 1)
        const float hs = fminf(fmaxf(y * (1.0f / 6.0f) + 0.5f, 0.0f), 1.0f);
        s[(size_t)mm * C_DIM + nn] = hs;
    }
}

// ---------------------------------------------------------------------------
// Kernel 3: out[b,c,:,:] = x[b,c,:,:] * s[b,c].  One block per (b,c) row.
// Non-temporal stores: output is never re-read, keep it out of L2.
// ---------------------------------------------------------------------------
__global__ void ese_scale_kernel(const float* __restrict__ x,
                                 const float* __restrict__ s,
                                 float* __restrict__ out) {
    const int row = blockIdx.x;                       // b*C + c
    const float sv = s[row];                          // uniform -> scalar load
    const v4f* xr = (const v4f*)(x + (size_t)row * HW);
    v4f* orow     = (v4f*)(out + (size_t)row * HW);

    for (int i = threadIdx.x; i < HW4; i += blockDim.x) {
        v4f v = xr[i];
        v *= sv;
        __builtin_nontemporal_store(v, &orow[i]);
    }
}

// ---------------------------------------------------------------------------
extern "C" void kernel_launch(void* const* d_in, const int* in_sizes, int n_in,
                              void* d_out, int out_size, void* d_ws, size_t ws_size,
                              hipStream_t stream) {
    const float* x    = (const float*)d_in[0];   // [32,768,56,56]
    const float* Wm   = (const float*)d_in[1];   // [768,768]
    const float* bias = (const float*)d_in[2];   // [768]
    float* out = (float*)d_out;

    float* xse = (float*)d_ws;                   // [32,768]
    float* s   = xse + (size_t)B_DIM * C_DIM;    // [32,768]

    // 1) squeeze: global average pool
    ese_pool_kernel<<<B_DIM * C_DIM, 256, 0, stream>>>(x, xse);

    // 2) excite: fp32 WMMA dense + bias + hardsigmoid (one wave per 16x16 tile)
    ese_gemm_hsig_kernel<<<dim3(C_DIM / 16, B_DIM / 16), 32, 0, stream>>>(
        xse, Wm, bias, s);

    // 3) scale
    ese_scale_kernel<<<B_DIM * C_DIM, 256, 0, stream>>>(x, s, out);
}